// TAdaConv3d_15444702396422
// MI455X (gfx1250) — compile-verified
//
#include <hip/hip_runtime.h>
#include <hip/hip_bf16.h>
#include <stdint.h>

typedef float v2f __attribute__((ext_vector_type(2)));
typedef float v8f __attribute__((ext_vector_type(8)));

#if defined(__gfx1250__) && __has_builtin(__builtin_amdgcn_global_load_async_to_lds_b32)
#define USE_ASYNC_LDS 1
typedef __attribute__((address_space(1))) int as1_int;
typedef __attribute__((address_space(3))) int as3_int;
#endif

#define CIN   256
#define COUT  256
#define TT    8
#define HH    56
#define WW    56
#define KCH   16          // C_in channels staged per LDS chunk
#define CPAD  18          // ci dim padded for conflict-free lane stride

// Implicit-GEMM 3x3 conv. alpha folded into WEIGHTS (per-(b,t) block), so the
// activation tile is a raw async DMA into LDS. Full-f32 WMMA (16x16x4).
// Block: 256 threads (8 waves). Tile: 64 co x (8 rows x 16 cols) pixels.
__global__ __launch_bounds__(256)
void tada_conv3d_wmma_f32(const float* __restrict__ x,
                          const float* __restrict__ alpha,
                          const float* __restrict__ wgt,
                          const float* __restrict__ bias,
                          float* __restrict__ out)
{
    // input tile with halo, channel-innermost: [row 0..9][col 0..17][ci]
    __shared__ float lds_x[10][18][CPAD];               // 12,960 B
    // weights as [pos][co_local][ci] (alpha-scaled)
    __shared__ float lds_w[9][64][CPAD];                // 41,472 B

    const int bt  = blockIdx.z;            // b*T + t
    const int b   = bt / TT;
    const int t   = bt % TT;
    const int co0 = blockIdx.y * 64;
    const int x0  = (blockIdx.x & 3)  * 16;   // 4 col tiles (last masked)
    const int y0  = (blockIdx.x >> 2) * 8;    // 7 row tiles

    const int tid  = threadIdx.x;
    const int wave = tid >> 5;
    const int lane = tid & 31;
    const int wm   = wave >> 1;            // 0..3 -> co quarter
    const int wn   = wave & 1;             // 0..1 -> 4-row half
    const int g    = lane >> 4;            // half-wave group
    const int ml   = lane & 15;

    v8f c0 = {}, c1 = {}, c2 = {}, c3 = {};

    // ---- zero the OOB halo cells once: position set is chunk-invariant ----
    for (int idx = tid; idx < 10 * 18; idx += 256) {
        int row = idx / 18, col = idx % 18;
        int yy = y0 - 1 + row, xx = x0 - 1 + col;
        if (yy < 0 || yy >= HH || xx < 0 || xx >= WW) {
            #pragma unroll
            for (int ci = 0; ci < KCH; ++ci) lds_x[row][col][ci] = 0.0f;
        }
    }

    const int ciw = tid & 15;              // this thread's fixed ci for weights

    for (int ci0 = 0; ci0 < CIN; ci0 += KCH) {
        __syncthreads();                   // previous chunk's LDS reads done

        // ---- 1) kick off async DMA of the raw input tile into LDS ----
        for (int idx = tid; idx < 10 * 18 * KCH; idx += 256) {
            int ci  = idx / 180;
            int rem = idx % 180;
            int row = rem / 18;
            int col = rem % 18;
            int yy  = y0 - 1 + row;
            int xx  = x0 - 1 + col;
            if (yy >= 0 && yy < HH && xx >= 0 && xx < WW) {
                const float* gp = x + (size_t)((b * CIN + ci0 + ci) * TT + t)
                                        * (HH * WW) + yy * WW + xx;
                float* lp = &lds_x[row][col][ci];
#if defined(USE_ASYNC_LDS)
                __builtin_amdgcn_global_load_async_to_lds_b32(
                    (as1_int*)(uintptr_t)gp,
                    (as3_int*)(uint32_t)(uintptr_t)lp, 0, 0);
#else
                *lp = *gp;
#endif
            }
        }

        // ---- 2) stage alpha-scaled weights (overlaps the async DMA) ----
        {
            const float av = alpha[(b * CIN + ci0 + ciw) * TT + t];
            #pragma unroll 4
            for (int it = 0; it < 36; ++it) {          // 9*64*16 / 256
                int idx = tid + (it << 8);
                int pos = idx >> 10;
                int co  = (idx >> 4) & 63;
                lds_w[pos][co][ciw] =
                    av * wgt[(size_t)(co0 + co) * (CIN * 9)
                             + (size_t)(ci0 + ciw) * 9 + pos];
            }
        }

        // prefetch next weight chunk toward L2 (global_prefetch_b8)
        if (ci0 + KCH < CIN) {
            const float* nxt = &wgt[(size_t)(co0 + (tid >> 2)) * (CIN * 9)
                                    + (ci0 + KCH) * 9 + (tid & 3)];
            __builtin_prefetch(nxt, 0, 2);
        }

        // ---- 3) wait for DMA, then sync the workgroup ----
#if defined(USE_ASYNC_LDS)
#if __has_builtin(__builtin_amdgcn_s_wait_asynccnt)
        __builtin_amdgcn_s_wait_asynccnt(0);
#else
        asm volatile("s_wait_asynccnt 0" ::: "memory");
#endif
#endif
        __syncthreads();

        // ---- 4) implicit GEMM: 9 taps x (KCH/4) f32-WMMA k-steps ----
        for (int pos = 0; pos < 9; ++pos) {
            const int kh = pos / 3;
            const int kw = pos % 3;
            const int rb = 4 * wn + kh;            // input row of N-tile 0
            const int cb = ml + kw;                // input col for this lane
            #pragma unroll
            for (int k0 = 0; k0 < KCH; k0 += 4) {
                const int kk = k0 + 2 * g;         // lane-group K pair base
                // A frag: lane<16 -> {K=k0,k0+1}, lane>=16 -> {k0+2,k0+3}
                v2f a = *(const v2f*)&lds_w[pos][wm * 16 + ml][kk];
                // B frags: same per-lane K arrangement, one b64 each
                v2f bf0 = *(const v2f*)&lds_x[rb    ][cb][kk];
                v2f bf1 = *(const v2f*)&lds_x[rb + 1][cb][kk];
                v2f bf2 = *(const v2f*)&lds_x[rb + 2][cb][kk];
                v2f bf3 = *(const v2f*)&lds_x[rb + 3][cb][kk];
                c0 = __builtin_amdgcn_wmma_f32_16x16x4_f32(
                        false, a, false, bf0, (short)0, c0, false, false);
                c1 = __builtin_amdgcn_wmma_f32_16x16x4_f32(
                        false, a, false, bf1, (short)0, c1, false, false);
                c2 = __builtin_amdgcn_wmma_f32_16x16x4_f32(
                        false, a, false, bf2, (short)0, c2, false, false);
                c3 = __builtin_amdgcn_wmma_f32_16x16x4_f32(
                        false, a, false, bf3, (short)0, c3, false, false);
            }
        }
    }

    // ---- epilogue: bias add + masked store ----
    const int xcol = x0 + ml;
    if (xcol < WW) {
        const int yb = y0 + 4 * wn;
        #pragma unroll
        for (int j = 0; j < 8; ++j) {
            int M  = j + 8 * g;                // C/D layout: lanes>=16 hold M+8
            int co = co0 + wm * 16 + M;
            float bv = bias[co];
            size_t base = ((size_t)(b * COUT + co) * TT + t) * (HH * WW);
            out[base + (size_t)(yb    ) * WW + xcol] = c0[j] + bv;
            out[base + (size_t)(yb + 1) * WW + xcol] = c1[j] + bv;
            out[base + (size_t)(yb + 2) * WW + xcol] = c2[j] + bv;
            out[base + (size_t)(yb + 3) * WW + xcol] = c3[j] + bv;
        }
    }
}

extern "C" void kernel_launch(void* const* d_in, const int* in_sizes, int n_in,
                              void* d_out, int out_size, void* d_ws, size_t ws_size,
                              hipStream_t stream) {
    const float* x     = (const float*)d_in[0];   // (4,256,8,56,56)
    const float* alpha = (const float*)d_in[1];   // (4,256,8,1,1)
    const float* wgt   = (const float*)d_in[2];   // (256,256,1,3,3)
    const float* bias  = (const float*)d_in[3];   // (256,)
    float* out = (float*)d_out;                   // (4,256,8,56,56)

    dim3 grid(4 * 7,       // 4 col tiles * 7 row tiles
              COUT / 64,   // c_out tiles
              4 * TT);     // b*t images
    tada_conv3d_wmma_f32<<<grid, 256, 0, stream>>>(x, alpha, wgt, bias, out);
}